// DrugEncoder_20658792694515
// MI455X (gfx1250) — compile-verified
//
#include <hip/hip_runtime.h>

// GINE-style GNN layer for MI455X (gfx1250, wave32).
// Pipeline: h = x@Wx+bx (WMMA f32)  ->  edge msg + atomic scatter-add
//           -> fused 3-layer MLP (WMMA f32, weights+acts in LDS)
//           -> edge output, recomputing e on the fly (never spilled to HBM).

#define N_NODES 100000
#define N_EDGES 1600000
#define IN_DIM  70
#define E_DIM   6
#define HID     64

#define KPAD 72   // IN_DIM padded to multiple of 4 (WMMA K step)
#define WSTR 72   // LDS stride for weight tiles: B-frag half-waves hit disjoint banks
#define ZSTR 73   // LDS stride for activation tiles: conflict-free A-frags (gcd(73,64)=1)

typedef float v2f __attribute__((ext_vector_type(2)));
typedef float v8f __attribute__((ext_vector_type(8)));

// ---------------------------------------------------------------------------
// 16x16 output tile GEMM for one wave32: C += A(16xK) * B(Kx16), fp32 WMMA.
// A: LDS, row stride ZSTR (pointer pre-offset to this wave's 16-row slab).
// B: LDS, row stride WSTR (full 64+ wide, nn selects this wave's 16 cols).
// ISA layout: lane L -> A[m=L&15][k+(L>>4)*2 .. +1]; B mirrored on K.
// ---------------------------------------------------------------------------
__device__ __forceinline__ v8f tile_gemm(const float* __restrict__ A,
                                         const float* __restrict__ B,
                                         int K, int m, int kb, int nn) {
  v8f acc = {};
#pragma unroll
  for (int k = 0; k < K; k += 4) {
    v2f a, b;
    a.x = A[m * ZSTR + k + kb];
    a.y = A[m * ZSTR + k + kb + 1];
    b.x = B[(k + kb) * WSTR + nn];
    b.y = B[(k + kb + 1) * WSTR + nn];
    acc = __builtin_amdgcn_wmma_f32_16x16x4_f32(false, a, false, b,
                                                (short)0, acc, false, false);
  }
  return acc;
}

// ---------------------------------------------------------------------------
// Zero the aggregation buffer (harness does not re-zero ws between replays).
// ---------------------------------------------------------------------------
__global__ void k_zero(float4* __restrict__ p, int n4) {
  int i = blockIdx.x * blockDim.x + threadIdx.x;
  if (i < n4) p[i] = make_float4(0.f, 0.f, 0.f, 0.f);
}

// ---------------------------------------------------------------------------
// h = x @ Wx + bx    [100000 x 70] @ [70 x 64]
// Block = 256 thr = 8 waves (2x4 tile grid) = 32 rows x 64 cols. Grid = 3125.
// x tile staged in LDS (coalesced load, zero-padded K), Wx staged in LDS.
// ---------------------------------------------------------------------------
__global__ __launch_bounds__(256)
void k_node_in(const float* __restrict__ x, const float* __restrict__ Wx,
               const float* __restrict__ bx, float* __restrict__ h) {
  __shared__ float Wl[KPAD * WSTR];
  __shared__ float Xl[32 * ZSTR];
  __shared__ float bl[HID];
  const int tid  = threadIdx.x;
  const int rowB = blockIdx.x * 32;

  for (int i = tid; i < KPAD * WSTR; i += 256) {
    int k = i / WSTR, n = i % WSTR;
    Wl[i] = (k < IN_DIM && n < HID) ? Wx[k * HID + n] : 0.f;
  }
  for (int i = tid; i < 32 * KPAD; i += 256) {
    int r = i / KPAD, kk = i % KPAD;
    Xl[r * ZSTR + kk] = (kk < IN_DIM) ? x[(rowB + r) * IN_DIM + kk] : 0.f;
  }
  if (tid < HID) bl[tid] = bx[tid];
  __syncthreads();

  const int lane = tid & 31, wid = tid >> 5;
  const int wm = wid >> 2, wn = wid & 3;
  const int n0 = wn * 16;
  const int m = lane & 15, kb = (lane >> 4) * 2, nn = n0 + (lane & 15);
  const int hrow = (lane >> 4) * 8;

  v8f acc = tile_gemm(&Xl[wm * 16 * ZSTR], Wl, KPAD, m, kb, nn);

#pragma unroll
  for (int g = 0; g < 8; ++g) {
    int r = rowB + wm * 16 + g + hrow;
    h[r * HID + nn] = acc[g] + bl[nn];
  }
}

// ---------------------------------------------------------------------------
// Edge message: e = (h[src]+h[dst] + ea@We+be)/3 ; msg = relu(h[src]+e)
// aggr[dst] += msg (L2-resident fp32 atomics). 4 edges/block, 64 lanes/edge.
// ---------------------------------------------------------------------------
__global__ __launch_bounds__(256)
void k_edge_msg(const float* __restrict__ eattr, const int* __restrict__ eidx,
                const float* __restrict__ We, const float* __restrict__ be,
                const float* __restrict__ h, float* __restrict__ aggr) {
  __shared__ float Wl[E_DIM * HID];
  __shared__ float bl[HID];
  const int tid = threadIdx.x;
  for (int i = tid; i < E_DIM * HID; i += 256) Wl[i] = We[i];
  if (tid < HID) bl[tid] = be[tid];
  __syncthreads();

  const int e = blockIdx.x * 4 + (tid >> 6);  // constant per wave
  const int c = tid & 63;
  const int src = eidx[e];
  const int dst = eidx[N_EDGES + e];

  float ec = bl[c];
#pragma unroll
  for (int j = 0; j < E_DIM; ++j)
    ec += eattr[e * E_DIM + j] * Wl[j * HID + c];

  const float hs = h[src * HID + c];
  const float hd = h[dst * HID + c];
  const float ev = (hs + hd + ec) * (1.0f / 3.0f);
  float msg = hs + ev;
  msg = msg > 0.f ? msg : 0.f;
  atomicAdd(&aggr[dst * HID + c], msg);
}

// ---------------------------------------------------------------------------
// Fused MLP: z = prelu(prelu((h+aggr)@W1+b1)@W2+b2)@W3+b3
// All three 64x64 weights in LDS; activations ping-pong Za<->Zb in LDS.
// Block = 32 rows; 8 waves each own a 16x16 tile per GEMM.
// ---------------------------------------------------------------------------
__global__ __launch_bounds__(256)
void k_mlp(const float* __restrict__ h, const float* __restrict__ aggr,
           const float* __restrict__ W1, const float* __restrict__ b1,
           const float* __restrict__ a1p,
           const float* __restrict__ W2, const float* __restrict__ b2,
           const float* __restrict__ a2p,
           const float* __restrict__ W3, const float* __restrict__ b3,
           float* __restrict__ z) {
  __shared__ float W1l[HID * WSTR], W2l[HID * WSTR], W3l[HID * WSTR];
  __shared__ float Za[32 * ZSTR], Zb[32 * ZSTR];
  __shared__ float b1l[HID], b2l[HID], b3l[HID];

  const int tid  = threadIdx.x;
  const int rowB = blockIdx.x * 32;

  for (int i = tid; i < HID * WSTR; i += 256) {
    int k = i / WSTR, n = i % WSTR;
    float w1 = 0.f, w2 = 0.f, w3 = 0.f;
    if (n < HID) { w1 = W1[k*HID+n]; w2 = W2[k*HID+n]; w3 = W3[k*HID+n]; }
    W1l[i] = w1; W2l[i] = w2; W3l[i] = w3;
  }
  if (tid < HID) { b1l[tid] = b1[tid]; b2l[tid] = b2[tid]; b3l[tid] = b3[tid]; }
  const float s1 = a1p[0], s2 = a2p[0];

  for (int i = tid; i < 32 * HID; i += 256) {
    int r = i >> 6, n = i & 63;
    int gi = (rowB + r) * HID + n;
    Za[r * ZSTR + n] = h[gi] + aggr[gi];
  }
  __syncthreads();

  const int lane = tid & 31, wid = tid >> 5;
  const int wm = wid >> 2, wn = wid & 3;
  const int n0 = wn * 16;
  const int m = lane & 15, kb = (lane >> 4) * 2, nn = n0 + (lane & 15);
  const int hrow = (lane >> 4) * 8;

  // GEMM1 + PReLU -> Zb
  v8f acc = tile_gemm(&Za[wm * 16 * ZSTR], W1l, HID, m, kb, nn);
#pragma unroll
  for (int g = 0; g < 8; ++g) {
    float v = acc[g] + b1l[nn];
    v = v >= 0.f ? v : s1 * v;
    Zb[(wm * 16 + g + hrow) * ZSTR + nn] = v;
  }
  __syncthreads();

  // GEMM2 + PReLU -> Za
  acc = tile_gemm(&Zb[wm * 16 * ZSTR], W2l, HID, m, kb, nn);
#pragma unroll
  for (int g = 0; g < 8; ++g) {
    float v = acc[g] + b2l[nn];
    v = v >= 0.f ? v : s2 * v;
    Za[(wm * 16 + g + hrow) * ZSTR + nn] = v;
  }
  __syncthreads();

  // GEMM3 -> z (global)
  acc = tile_gemm(&Za[wm * 16 * ZSTR], W3l, HID, m, kb, nn);
#pragma unroll
  for (int g = 0; g < 8; ++g) {
    int r = rowB + wm * 16 + g + hrow;
    z[r * HID + nn] = acc[g] + b3l[nn];
  }
}

// ---------------------------------------------------------------------------
// e_out = (z[src]+z[dst] + e)/3, with e recomputed (h stays L2-resident,
// edge_attr re-read is 38 MB vs 820 MB spill/reload of e). NT store: e_out
// is 410 MB streaming, keep it out of L2.
// ---------------------------------------------------------------------------
__global__ __launch_bounds__(256)
void k_edge_out(const float* __restrict__ eattr, const int* __restrict__ eidx,
                const float* __restrict__ We, const float* __restrict__ be,
                const float* __restrict__ h, const float* __restrict__ z,
                float* __restrict__ eout) {
  __shared__ float Wl[E_DIM * HID];
  __shared__ float bl[HID];
  const int tid = threadIdx.x;
  for (int i = tid; i < E_DIM * HID; i += 256) Wl[i] = We[i];
  if (tid < HID) bl[tid] = be[tid];
  __syncthreads();

  const int e = blockIdx.x * 4 + (tid >> 6);
  const int c = tid & 63;
  const int src = eidx[e];
  const int dst = eidx[N_EDGES + e];

  float ec = bl[c];
#pragma unroll
  for (int j = 0; j < E_DIM; ++j)
    ec += eattr[e * E_DIM + j] * Wl[j * HID + c];

  const float hs = h[src * HID + c];
  const float hd = h[dst * HID + c];
  const float ev = (hs + hd + ec) * (1.0f / 3.0f);

  const float zs = z[src * HID + c];
  const float zd = z[dst * HID + c];
  const float r  = (zs + zd + ev) * (1.0f / 3.0f);
  __builtin_nontemporal_store(r, &eout[(size_t)e * HID + c]);
}

// ---------------------------------------------------------------------------
extern "C" void kernel_launch(void* const* d_in, const int* in_sizes, int n_in,
                              void* d_out, int out_size, void* d_ws, size_t ws_size,
                              hipStream_t stream) {
  const float* x     = (const float*)d_in[0];
  const float* eattr = (const float*)d_in[1];
  const int*   eidx  = (const int*)d_in[2];   // [2, E] flat: row0 = src, row1 = dst
  const float* Wx = (const float*)d_in[3];
  const float* bx = (const float*)d_in[4];
  const float* We = (const float*)d_in[5];
  const float* be = (const float*)d_in[6];
  const float* W1 = (const float*)d_in[7];
  const float* b1 = (const float*)d_in[8];
  const float* a1 = (const float*)d_in[9];
  const float* W2 = (const float*)d_in[10];
  const float* b2 = (const float*)d_in[11];
  const float* a2 = (const float*)d_in[12];
  const float* W3 = (const float*)d_in[13];
  const float* b3 = (const float*)d_in[14];

  float* z    = (float*)d_out;                         // [N_NODES, HID]
  float* eout = (float*)d_out + (size_t)N_NODES * HID; // [N_EDGES, HID]

  float* h    = (float*)d_ws;                          // 25.6 MB
  float* aggr = h + (size_t)N_NODES * HID;             // 25.6 MB

  const int n4 = N_NODES * HID / 4;
  k_zero   <<<(n4 + 255) / 256, 256, 0, stream>>>((float4*)aggr, n4);
  k_node_in<<<N_NODES / 32, 256, 0, stream>>>(x, Wx, bx, h);
  k_edge_msg<<<N_EDGES / 4, 256, 0, stream>>>(eattr, eidx, We, be, h, aggr);
  k_mlp    <<<N_NODES / 32, 256, 0, stream>>>(h, aggr, W1, b1, a1, W2, b2, a2, W3, b3, z);
  k_edge_out<<<N_EDGES / 4, 256, 0, stream>>>(eattr, eidx, We, be, h, z, eout);
}